// SindyModel_8753143349705
// MI455X (gfx1250) — compile-verified
//
#include <hip/hip_runtime.h>
#include <math.h>

typedef __attribute__((ext_vector_type(2))) float v2f;
typedef __attribute__((ext_vector_type(8))) float v8f;

#define POLY_L 23   // 20 polynomial + 3 sin columns
#define WAVES_PER_BLOCK 8

// Compute the 23 SINDy library features (sklearn combinations_with_replacement
// ordering: 1, x0,x1,x2, x0^2,x0x1,x0x2,x1^2,x1x2,x2^2, 10 cubics, sin(x0..2))
// and pack them directly into the 6 B-matrix register pairs for the
// V_WMMA_F32_16X16X4_F32 4x16 B layout:
//   chunk c, VGPR0: lanes 0-15 = K=4c,   lanes 16-31 = K=4c+2
//   chunk c, VGPR1: lanes 0-15 = K=4c+1, lanes 16-31 = K=4c+3
// All selects are between NAMED scalars (one v_cndmask each) — never array
// elements, which clang would turn into a dynamic-index cndmask tree.
__device__ __forceinline__ void theta_pack(float x0, float x1, float x2,
                                           bool hi, v2f B[6]) {
    const float x00 = x0 * x0, x01 = x0 * x1, x02 = x0 * x2;
    const float x11 = x1 * x1, x12 = x1 * x2, x22 = x2 * x2;
    const float c000 = x00 * x0;   // x0^3      (K=10)
    const float c001 = x00 * x1;   // x0^2 x1   (K=11)
    const float c002 = x00 * x2;   // x0^2 x2   (K=12)
    const float c011 = x0 * x11;   // x0 x1^2   (K=13)
    const float c012 = x01 * x2;   // x0 x1 x2  (K=14)
    const float c022 = x0 * x22;   // x0 x2^2   (K=15)
    const float c111 = x11 * x1;   // x1^3      (K=16)
    const float c112 = x11 * x2;   // x1^2 x2   (K=17)
    const float c122 = x22 * x1;   // x1 x2^2   (K=18)
    const float c222 = x22 * x2;   // x2^3      (K=19)
    const float s0 = __sinf(x0), s1 = __sinf(x1), s2 = __sinf(x2);

    // chunk 0: K 0..3  = {1, x0, x1, x2}
    B[0].x = hi ? x1   : 1.0f;   B[0].y = hi ? x2   : x0;
    // chunk 1: K 4..7  = {x0^2, x0x1, x0x2, x1^2}
    B[1].x = hi ? x02  : x00;    B[1].y = hi ? x11  : x01;
    // chunk 2: K 8..11 = {x1x2, x2^2, x0^3, x0^2x1}
    B[2].x = hi ? c000 : x12;    B[2].y = hi ? c001 : x22;
    // chunk 3: K 12..15 = {x0^2x2, x0x1^2, x0x1x2, x0x2^2}
    B[3].x = hi ? c012 : c002;   B[3].y = hi ? c022 : c011;
    // chunk 4: K 16..19 = {x1^3, x1^2x2, x1x2^2, x2^3}
    B[4].x = hi ? c122 : c111;   B[4].y = hi ? c222 : c112;
    // chunk 5: K 20..23 = {sin x0, sin x1, sin x2, 0}
    B[5].x = hi ? s2   : s0;     B[5].y = hi ? 0.0f : s1;
}

// Chain 6 x V_WMMA_F32_16X16X4_F32 over K=24.
__device__ __forceinline__ v8f project_wmma(const v2f A[6], const v2f B[6],
                                            v8f acc) {
#pragma unroll
    for (int c = 0; c < 6; ++c) {
        acc = __builtin_amdgcn_wmma_f32_16x16x4_f32(
            /*neg_a=*/false, A[c], /*neg_b=*/false, B[c],
            /*c_mod=*/(short)0, acc, /*reuse_a=*/false, /*reuse_b=*/false);
    }
    return acc;
}

__global__ __launch_bounds__(32 * WAVES_PER_BLOCK)
void sindy_wmma_kernel(const float* __restrict__ x,
                       const float* __restrict__ W,
                       const float* __restrict__ mask,
                       float* __restrict__ out,
                       long long P /* number of points */) {
    const int  lane  = threadIdx.x & 31;
    const int  row   = lane & 15;       // A-matrix row (= output column s) / point slot
    const bool hi    = lane >= 16;      // upper lane half -> K+2..K+3 slice
    const int  khalf = hi ? 2 : 0;

    // Preload A = (mask*W)^T padded to 16x24 in the 32-bit 16x4 A layout:
    // lanes 0-15 hold K=4c..4c+1, lanes 16-31 hold K=4c+2..4c+3, row = M.
    v2f A[6];
#pragma unroll
    for (int c = 0; c < 6; ++c) {
#pragma unroll
        for (int j = 0; j < 2; ++j) {
            const int l = 4 * c + khalf + j;   // library column index (K)
            float v = 0.0f;
            if (row < 3 && l < POLY_L)
                v = W[l * 3 + row] * mask[l * 3 + row];
            A[c][j] = v;
        }
    }

    const long long G      = (P + 15) >> 4;    // 16-point groups
    const long long nwaves = (long long)gridDim.x * WAVES_PER_BLOCK;
    const long long wid    = ((long long)blockIdx.x * blockDim.x + threadIdx.x) >> 5;
    const long long gstep  = nwaves * 2;

    for (long long g = wid * 2; g < G; g += gstep) {
        // Prefetch next iteration's input lines (speculative; OOB is dropped).
        __builtin_prefetch(x + (g + gstep) * 48, 0, 0);

        // ---- group 0 ----
        const long long p0  = g * 16 + row;
        const long long pc0 = (p0 < P) ? p0 : (P - 1);  // clamp keeps EXEC all-1s
        const float* xp0 = x + pc0 * 3;
        v2f B0[6];
        theta_pack(__builtin_nontemporal_load(xp0 + 0),
                   __builtin_nontemporal_load(xp0 + 1),
                   __builtin_nontemporal_load(xp0 + 2), hi, B0);

        // ---- group 1 ----
        const long long p1  = (g + 1) * 16 + row;
        const long long pc1 = (p1 < P) ? p1 : (P - 1);
        const float* xp1 = x + pc1 * 3;
        v2f B1[6];
        theta_pack(__builtin_nontemporal_load(xp1 + 0),
                   __builtin_nontemporal_load(xp1 + 1),
                   __builtin_nontemporal_load(xp1 + 2), hi, B1);

        // Two independent WMMA chains -> ILP on the matrix pipe.
        v8f acc0 = {};
        v8f acc1 = {};
        acc0 = project_wmma(A, B0, acc0);
        acc1 = project_wmma(A, B1, acc1);

        // D layout: VGPR r, lanes 0-15 -> out[point = lane][s = r], r = 0..2.
        if (!hi) {
            if (p0 < P) {
                float* op = out + p0 * 3;
                __builtin_nontemporal_store(acc0[0], op + 0);
                __builtin_nontemporal_store(acc0[1], op + 1);
                __builtin_nontemporal_store(acc0[2], op + 2);
            }
            if (p1 < P) {
                float* op = out + p1 * 3;
                __builtin_nontemporal_store(acc1[0], op + 0);
                __builtin_nontemporal_store(acc1[1], op + 1);
                __builtin_nontemporal_store(acc1[2], op + 2);
            }
        }
    }
}

extern "C" void kernel_launch(void* const* d_in, const int* in_sizes, int n_in,
                              void* d_out, int out_size, void* d_ws, size_t ws_size,
                              hipStream_t stream) {
    const float* x    = (const float*)d_in[0];   // [B, N, 3] f32
    const float* W    = (const float*)d_in[1];   // [23, 3]  f32
    const float* mask = (const float*)d_in[2];   // [23, 3]  f32
    float*       out  = (float*)d_out;           // [B, N, 3] f32

    const long long P = (long long)in_sizes[0] / 3;   // 6.4M points
    const long long G = (P + 15) >> 4;                // 16-point groups

    const int threads = 32 * WAVES_PER_BLOCK;         // 256 = 8 waves
    long long wavePairs = (G + 1) / 2;                // 2 groups per wave per sweep
    long long blocksLL  = (wavePairs + WAVES_PER_BLOCK - 1) / WAVES_PER_BLOCK;
    if (blocksLL > 4096) blocksLL = 4096;             // persistent waves; grid-stride loop
    if (blocksLL < 1)    blocksLL = 1;

    sindy_wmma_kernel<<<(int)blocksLL, threads, 0, stream>>>(x, W, mask, out, P);
}